// DynamicChannelSpecificPatches_33809982554196
// MI455X (gfx1250) — compile-verified
//
#include <hip/hip_runtime.h>

typedef __attribute__((ext_vector_type(2))) float v2f;
typedef __attribute__((ext_vector_type(8))) float v8f;

#define IMG    224
#define PATCH  16
#define NP     14            // patches per side
#define EMBED  768
#define CCH    12
#define BB     64
#define M_TOTAL (BB * NP * NP)   // 12544
#define NTILES  (EMBED / 16)     // 48
#define MTILES  (M_TOTAL / 16)   // 784

// ---------------------------------------------------------------------------
// Stage 1: 16x16 sum-pool. One block per (b, c, patch-row) strip of 16 rows.
// 224 threads (7 waves): thread t sums column t over 16 rows (fully coalesced
// 896B row reads), then 14 threads reduce 16 columns each -> patch sums.
// Output layout: pooled[m * 12 + c], m = b*196 + i*14 + j  (row-major A matrix)
// ---------------------------------------------------------------------------
__global__ __launch_bounds__(224) void pool16_kernel(const float* __restrict__ x,
                                                     float* __restrict__ pooled) {
    __shared__ float colsum[IMG];
    int s = blockIdx.x;              // 0 .. B*C*NP-1
    int i = s % NP; s /= NP;         // patch row
    int c = s % CCH; s /= CCH;       // channel
    int b = s;                       // batch
    const float* base = x + (((size_t)(b * CCH + c)) * IMG + (size_t)i * PATCH) * IMG;

    int col = threadIdx.x;
    float acc = 0.f;
#pragma unroll
    for (int r = 0; r < PATCH; ++r)
        acc += base[(size_t)r * IMG + col];
    colsum[col] = acc;
    __syncthreads();

    if (col < NP) {
        float ps = 0.f;
#pragma unroll
        for (int k = 0; k < PATCH; ++k)
            ps += colsum[col * PATCH + k];
        int m = b * (NP * NP) + i * NP + col;
        pooled[(size_t)m * CCH + c] = ps;
    }
}

// ---------------------------------------------------------------------------
// Stage 2: out[M=12544, N=768] = pooled[M,12] x W[12,768] via
// V_WMMA_F32_16X16X4_F32, K=12 -> 3 chained WMMAs per 16x16 tile.
// One wave per tile, 8 waves per 256-thread block.
// ---------------------------------------------------------------------------
__global__ __launch_bounds__(256) void gemm_wmma_kernel(const float* __restrict__ pooled,
                                                        const float* __restrict__ w,
                                                        float* __restrict__ out) {
    int wid  = blockIdx.x * 8 + (threadIdx.x >> 5);
    int lane = threadIdx.x & 31;
    int mtile = wid / NTILES;
    int ntile = wid % NTILES;

    int half  = lane >> 4;           // 0 for lanes 0-15, 1 for lanes 16-31
    int khalf = half * 2;            // K offset of this lane-half
    int mrow  = lane & 15;           // A: M index
    int n     = lane & 15;           // B/D: N index

    const float* arow = pooled + (size_t)(mtile * 16 + mrow) * CCH;
    const float* wcol = w + ntile * 16 + n;

    v8f acc = {};                    // C = 0
#pragma unroll
    for (int k0 = 0; k0 < CCH; k0 += 4) {
        v2f a;                       // A 16x4: VGPR0 = K=k0+khalf, VGPR1 = +1
        a.x = arow[k0 + khalf];
        a.y = arow[k0 + khalf + 1];
        v2f bm;                      // B 4x16: same K split across lane halves
        bm.x = wcol[(size_t)(k0 + khalf)     * EMBED];
        bm.y = wcol[(size_t)(k0 + khalf + 1) * EMBED];
        acc = __builtin_amdgcn_wmma_f32_16x16x4_f32(
            /*neg_a=*/false, a, /*neg_b=*/false, bm,
            /*c_mod=*/(short)0, acc, /*reuse_a=*/false, /*reuse_b=*/false);
    }

    // D layout: VGPR v -> M = v + 8*half, N = lane&15
    int mbase = mtile * 16 + half * 8;
    float* o = out + (size_t)mbase * EMBED + (size_t)ntile * 16 + n;
#pragma unroll
    for (int v = 0; v < 8; ++v)
        o[(size_t)v * EMBED] = acc[v];
}

extern "C" void kernel_launch(void* const* d_in, const int* in_sizes, int n_in,
                              void* d_out, int out_size, void* d_ws, size_t ws_size,
                              hipStream_t stream) {
    const float* x = (const float*)d_in[0];   // [64, 12, 224, 224] fp32
    const float* w = (const float*)d_in[1];   // [12, 768] fp32
    float* out    = (float*)d_out;            // [64, 196, 768] fp32
    float* pooled = (float*)d_ws;             // [12544, 12] fp32 (~602 KB)

    pool16_kernel<<<BB * CCH * NP, 224, 0, stream>>>(x, pooled);
    gemm_wmma_kernel<<<(MTILES * NTILES) / 8, 256, 0, stream>>>(pooled, w, out);
}